// Slayer_16655883173896
// MI455X (gfx1250) — compile-verified
//
#include <hip/hip_runtime.h>
#include <hip/hip_bf16.h>

typedef __attribute__((ext_vector_type(2))) float v2f;
typedef __attribute__((ext_vector_type(8))) float v8f;

#define IN_DIM   768
#define OUT_DIM  768
#define BATCH    32
#define NUM_F    8
#define KSIN     (IN_DIM * NUM_F)      // 6144
#define OUT_ELEMS (BATCH * OUT_DIM)    // 24576

// ws layout (floats):
//  [0 .. 196607]        Ssin_T[k][b]   k = i*8+f, 6144 x 32
//  [196608 .. 221183]   Sbase_T[i][b]  768 x 32
//  [221184 .. 442367]   partials[ks][o][b], 9 x 768 x 32
#define WS_SIN   0
#define WS_BASE  196608
#define WS_PART  221184

// ---------------------------------------------------------------------------
// Prep: build sin-basis and silu matrices, transposed to [k][batch] so the
// WMMA B-operand gathers are simple strided reads from a tiny L2-resident
// buffer. One thread per (i,b); lane index == b gives coalesced 128B stores.
// ---------------------------------------------------------------------------
__global__ void slayer_prep(const float* __restrict__ x,
                            const float* __restrict__ grid,
                            float* __restrict__ ws) {
    int tid = blockIdx.x * blockDim.x + threadIdx.x;   // 0 .. 24575
    if (tid >= IN_DIM * BATCH) return;
    int b = tid & 31;
    int i = tid >> 5;
    float xv = x[b * IN_DIM + i];
    float sig = 1.0f / (1.0f + __expf(-xv));
    ws[WS_BASE + i * 32 + b] = xv * sig;               // silu
    float* sinp = ws + WS_SIN + (i * NUM_F) * 32 + b;
#pragma unroll
    for (int f = 0; f < NUM_F; ++f) {
        sinp[f * 32] = __sinf(grid[f] * xv);
    }
}

// ---------------------------------------------------------------------------
// GEMM: D(16 outs x 16 batch) += A(16x4 f32) x B(4x16 f32) via
// v_wmma_f32_16x16x4_f32. One wave per block. blockIdx.x = output tile (48),
// blockIdx.y = K split: 0..7 stream 768-wide chunks of the scale_sp*coef
// matrix (contiguous per output row); 8 handles the scale_base * silu term.
// A-operand lane layout (ISA 7.12.2): lane<16 -> row=lane, K=k,k+1;
// lane>=16 -> row=lane-16, K=k+2,k+3.  B 4x16: vgpr0 lanes0-15 K=k,
// lanes16-31 K=k+2; vgpr1 K=k+1 / k+3.
// ---------------------------------------------------------------------------
__global__ __launch_bounds__(32)
void slayer_gemm(const float* __restrict__ coef,
                 const float* __restrict__ scale_base,
                 const float* __restrict__ scale_sp,
                 const float* __restrict__ ws_in,     // sin/base tables
                 float* __restrict__ ws_part) {
    const int lane  = threadIdx.x;        // 0..31
    const int half  = lane >> 4;          // 0 or 1
    const int l15   = lane & 15;
    const int o0    = blockIdx.x * 16;    // output-channel tile base
    const int ks    = blockIdx.y;         // 0..8
    const int o_row = o0 + l15;           // this lane's A row

    v8f c0 = {};  // batch 0..15
    v8f c1 = {};  // batch 16..31

    if (ks < 8) {
        // sin-basis chunk: K window [ks*768, ks*768+768) of 6144
        const float* arow  = coef + (size_t)o_row * KSIN + (size_t)ks * 768;
        const float* sprow = scale_sp + (size_t)o_row * IN_DIM + ks * 96; // (ks*768)>>3
        const float* Bb    = ws_in + WS_SIN + (size_t)ks * 768 * 32;
        for (int k = 0; k < 768; k += 4) {
            const int ka = k + half * 2;                   // this lane's K pair base
            v2f a = *(const v2f*)(arow + ka);              // coef row, contiguous stream
            __builtin_prefetch(arow + ka + 128, 0, 0);     // global_prefetch_b8, ~512B ahead
            const float sp = sprow[ka >> 3];               // fold scale_sp (same i for both)
            a.x *= sp; a.y *= sp;
            v2f b0, b1;
            b0.x = Bb[(ka    ) * 32 + l15];
            b0.y = Bb[(ka + 1) * 32 + l15];
            b1.x = Bb[(ka    ) * 32 + l15 + 16];
            b1.y = Bb[(ka + 1) * 32 + l15 + 16];
            c0 = __builtin_amdgcn_wmma_f32_16x16x4_f32(false, a, false, b0,
                                                       (short)0, c0, false, false);
            c1 = __builtin_amdgcn_wmma_f32_16x16x4_f32(false, a, false, b1,
                                                       (short)0, c1, false, false);
        }
    } else {
        // scale_base * silu(x) term: K = 768
        const float* arow = scale_base + (size_t)o_row * IN_DIM;
        const float* Bb   = ws_in + WS_BASE;
        for (int k = 0; k < 768; k += 4) {
            const int ka = k + half * 2;
            v2f a = *(const v2f*)(arow + ka);
            v2f b0, b1;
            b0.x = Bb[(ka    ) * 32 + l15];
            b0.y = Bb[(ka + 1) * 32 + l15];
            b1.x = Bb[(ka    ) * 32 + l15 + 16];
            b1.y = Bb[(ka + 1) * 32 + l15 + 16];
            c0 = __builtin_amdgcn_wmma_f32_16x16x4_f32(false, a, false, b0,
                                                       (short)0, c0, false, false);
            c1 = __builtin_amdgcn_wmma_f32_16x16x4_f32(false, a, false, b1,
                                                       (short)0, c1, false, false);
        }
    }

    // D layout (ISA 7.12.2): vgpr r, lanes0-15 -> M=r, lanes16-31 -> M=r+8.
    float* outp = ws_part + (size_t)ks * OUT_ELEMS;
#pragma unroll
    for (int r = 0; r < 8; ++r) {
        const int o = o0 + r + half * 8;
        outp[o * 32 + l15]      = c0[r];
        outp[o * 32 + l15 + 16] = c1[r];
    }
}

// ---------------------------------------------------------------------------
// Reduce the 9 K-split partials + bias into y[b, o]. Deterministic (fixed
// summation order) so graph replays are bit-identical.
// ---------------------------------------------------------------------------
__global__ void slayer_reduce(const float* __restrict__ bias_w,
                              const float* __restrict__ ws_part,
                              float* __restrict__ out) {
    int tid = blockIdx.x * blockDim.x + threadIdx.x;   // 0 .. 24575
    if (tid >= OUT_ELEMS) return;
    int o = tid % OUT_DIM;
    int b = tid / OUT_DIM;
    float acc = bias_w[o];
#pragma unroll
    for (int ks = 0; ks < 9; ++ks)
        acc += ws_part[(size_t)ks * OUT_ELEMS + o * 32 + b];
    out[tid] = acc;                                    // tid == b*OUT_DIM + o
}

extern "C" void kernel_launch(void* const* d_in, const int* in_sizes, int n_in,
                              void* d_out, int out_size, void* d_ws, size_t ws_size,
                              hipStream_t stream) {
    const float* x          = (const float*)d_in[0];   // [32,12,64] = [32,768]
    const float* grid       = (const float*)d_in[1];   // [8]
    const float* coef       = (const float*)d_in[2];   // [768*768, 8]
    const float* bias_w     = (const float*)d_in[3];   // [1,768]
    const float* scale_base = (const float*)d_in[4];   // [768*768]
    const float* scale_sp   = (const float*)d_in[5];   // [768*768]
    float* out = (float*)d_out;
    float* ws  = (float*)d_ws;

    // 1) sin/silu basis tables (24576 threads)
    slayer_prep<<<96, 256, 0, stream>>>(x, grid, ws);

    // 2) WMMA GEMM: 48 output tiles x 9 K-splits, one wave each
    slayer_gemm<<<dim3(48, 9), 32, 0, stream>>>(coef, scale_base, scale_sp,
                                                ws, ws + WS_PART);

    // 3) reduce partials + bias
    slayer_reduce<<<96, 256, 0, stream>>>(bias_w, ws + WS_PART, out);
}